// GCN_26792005992481
// MI455X (gfx1250) — compile-verified
//
#include <hip/hip_runtime.h>

#define NNODES 100000
#define NREL   8
#define NEDGE  1000000
#define EMBD   64
#define RN     (NREL * NNODES)

typedef __attribute__((ext_vector_type(2))) float v2f;
typedef __attribute__((ext_vector_type(8))) float v8f;

// ---------------------------------------------------------------------------
// Kernel 1: row-degree accumulation. deg[row] += 1 per edge.
// ---------------------------------------------------------------------------
__global__ void rgcn_degree(const int* __restrict__ rows, float* __restrict__ deg) {
    int e = blockIdx.x * blockDim.x + threadIdx.x;
    if (e < NEDGE) {
        atomicAdd(&deg[rows[e]], 1.0f);
    }
}

// ---------------------------------------------------------------------------
// Kernel 2: un-normalized SpMM scatter. One wave per edge; each lane handles
// two consecutive features (float2 load from x, two f32 atomics into hsum).
// Normalization by 1/deg is deferred to the dense kernel (linear scaling).
// x (25.6 MB) stays L2-resident; atomics stream into the 205 MB hsum.
// ---------------------------------------------------------------------------
__global__ void rgcn_scatter(const float* __restrict__ x,
                             const int*   __restrict__ rows,
                             const int*   __restrict__ cols,
                             float*       __restrict__ hsum) {
    int tid  = blockIdx.x * blockDim.x + threadIdx.x;
    int e    = tid >> 5;          // 32 lanes per edge
    int lane = tid & 31;
    if (e >= NEDGE) return;
    int row = rows[e];            // wave-uniform
    int col = cols[e];            // wave-uniform
    const float2 v = *(const float2*)(x + (size_t)col * EMBD + lane * 2);
    float* hr = hsum + (size_t)row * EMBD + lane * 2;
    atomicAdd(hr + 0, v.x);
    atomicAdd(hr + 1, v.y);
}

// ---------------------------------------------------------------------------
// Kernel 3: dense contraction out[n,f] = relu( sum_r (h[r,n,:]/deg) @ W[r] )
// via V_WMMA_F32_16X16X4_F32. One wave owns a 16-node x 64-feature strip with
// 4 accumulators; A (h) streams from global (each element read once per wave).
//
// B (W) is staged per-relation into double-buffered LDS with a K-PAIR
// INTERLEAVED layout:  lds[(k/2)*128 + n*2 + (k&1)] = W[k][n]
// so each WMMA B fragment (W[k][n], W[k+1][n]) is a contiguous 8-byte float2
// -> one ds_load_b64 straight into an even-aligned VGPR pair, no repacking
// movs, and all 64 B loads per relation use one base VGPR + immediate offsets.
// ---------------------------------------------------------------------------
__global__ void __launch_bounds__(256)
rgcn_dense_wmma(const float* __restrict__ hsum,
                const float* __restrict__ deg,
                const float* __restrict__ W,
                float*       __restrict__ out) {
    __shared__ float ldsW[2][EMBD * EMBD];    // 2 x 16 KB double buffer

    const int lane = threadIdx.x & 31;
    const int wave = (blockIdx.x * blockDim.x + threadIdx.x) >> 5;
    const bool active = (wave < NNODES / 16);  // wave-uniform; barriers still run

    const int m0   = wave * 16;               // node block base
    const int r16  = lane & 15;
    const int hi   = lane >> 4;               // lane half selects K pair
    const int node = m0 + r16;

    v8f acc0 = {}, acc1 = {}, acc2 = {}, acc3 = {};

    // Cooperative K-pair-interleaved staging of one relation's 64x64 tile.
    // Item i in [0,1024): p = K-pair index (0..31), nn = column-pair (0..31).
    // lds[p*128 + 4*nn .. +3] = { W[2p][2nn], W[2p+1][2nn],
    //                             W[2p][2nn+1], W[2p+1][2nn+1] }
    auto stage = [&](int rel, int buf) {
        const float* wsrc = W + (size_t)rel * EMBD * EMBD;
        float4* dst = (float4*)ldsW[buf];
#pragma unroll
        for (int it = 0; it < 1024 / 256; ++it) {
            int i  = threadIdx.x + it * 256;
            int p  = i >> 5;
            int nn = i & 31;
            const float2 g0 = *(const float2*)(wsrc + (2 * p)     * EMBD + 2 * nn);
            const float2 g1 = *(const float2*)(wsrc + (2 * p + 1) * EMBD + 2 * nn);
            dst[p * 32 + nn] = make_float4(g0.x, g1.x, g0.y, g1.y);
        }
    };

    stage(0, 0);

    for (int r = 0; r < NREL; ++r) {
        __syncthreads();   // buf[r&1] staged; previous iteration's compute done

        if (r + 1 < NREL) stage(r + 1, (r + 1) & 1);

        if (active) {
            float d    = deg[r * NNODES + node];
            float invd = (d > 0.0f) ? __builtin_amdgcn_rcpf(d) : 0.0f;
            const float* hrow = hsum + ((size_t)r * NNODES + node) * EMBD;
            // Per-lane base into interleaved tile: K-pair parity from lane
            // half (hi), column from r16; per-(kb,tile) deltas are immediates.
            const float* wbase = ldsW[r & 1] + (size_t)hi * 128 + (size_t)r16 * 2;
#pragma unroll
            for (int kb = 0; kb < EMBD / 4; ++kb) {
                const int k = kb * 4 + hi * 2;
                v2f a;
                a.x = hrow[k]     * invd;
                a.y = hrow[k + 1] * invd;

                const float* wk = wbase + kb * 256;           // kb*2 K-pairs
                v2f b0 = *(const v2f*)(wk + 0);
                v2f b1 = *(const v2f*)(wk + 32);              // tile n+16
                v2f b2 = *(const v2f*)(wk + 64);              // tile n+32
                v2f b3 = *(const v2f*)(wk + 96);              // tile n+48

                acc0 = __builtin_amdgcn_wmma_f32_16x16x4_f32(false, a, false, b0, (short)0, acc0, false, false);
                acc1 = __builtin_amdgcn_wmma_f32_16x16x4_f32(false, a, false, b1, (short)0, acc1, false, false);
                acc2 = __builtin_amdgcn_wmma_f32_16x16x4_f32(false, a, false, b2, (short)0, acc2, false, false);
                acc3 = __builtin_amdgcn_wmma_f32_16x16x4_f32(false, a, false, b3, (short)0, acc3, false, false);
            }
        }
    }

    if (!active) return;

    // Fused ReLU + store (C/D layout: row M = v + 8*hi, col N = tile*16 + r16)
    float* o = out + (size_t)(m0 + hi * 8) * EMBD + r16;
#pragma unroll
    for (int v = 0; v < 8; ++v) {
        float* orow = o + (size_t)v * EMBD;
        float t0 = acc0[v]; orow[0]  = t0 > 0.0f ? t0 : 0.0f;
        float t1 = acc1[v]; orow[16] = t1 > 0.0f ? t1 : 0.0f;
        float t2 = acc2[v]; orow[32] = t2 > 0.0f ? t2 : 0.0f;
        float t3 = acc3[v]; orow[48] = t3 > 0.0f ? t3 : 0.0f;
    }
}

// ---------------------------------------------------------------------------
// Launch: memset scratch -> degree -> scatter -> WMMA dense
// Workspace layout: [ deg : RN floats ][ hsum : RN*EMBD floats ]  (~208 MB)
// ---------------------------------------------------------------------------
extern "C" void kernel_launch(void* const* d_in, const int* in_sizes, int n_in,
                              void* d_out, int out_size, void* d_ws, size_t ws_size,
                              hipStream_t stream) {
    const float* x        = (const float*)d_in[0];   // [N, 64]
    const float* weights  = (const float*)d_in[1];   // [R, 64, 64]
    const int*   edge_row = (const int*)d_in[2];     // [E]
    const int*   edge_col = (const int*)d_in[3];     // [E]
    float*       out      = (float*)d_out;           // [N, 64]

    float* deg  = (float*)d_ws;
    float* hsum = deg + (size_t)RN;

    const size_t zero_bytes = (size_t)RN * (1 + EMBD) * sizeof(float);
    hipMemsetAsync(d_ws, 0, zero_bytes, stream);

    {   // degree: 1 thread / edge
        int threads = 256;
        int blocks  = (NEDGE + threads - 1) / threads;
        rgcn_degree<<<blocks, threads, 0, stream>>>(edge_row, deg);
    }
    {   // scatter: 1 wave / edge
        int threads = 256;
        long long total = (long long)NEDGE * 32;
        int blocks = (int)((total + threads - 1) / threads);
        rgcn_scatter<<<blocks, threads, 0, stream>>>(x, edge_row, edge_col, hsum);
    }
    {   // dense WMMA: 1 wave / 16-node strip -> 6250 waves (8 waves/block)
        int threads = 256;
        int waves   = NNODES / 16;               // 6250
        int blocks  = (waves * 32 + threads - 1) / threads;   // 782
        rgcn_dense_wmma<<<blocks, threads, 0, stream>>>(hsum, deg, weights, out);
    }
}